// RetrainUtils_14250701488865
// MI455X (gfx1250) — compile-verified
//
#include <hip/hip_runtime.h>
#include <stdint.h>

// Fixed anchor geometry from the reference:
//   level 0: 68x120 stride 8  -> anchors [0, 8160)
//   level 1: 34x60  stride 16 -> anchors [8160, 10200)
//   level 2: 17x30  stride 32 -> anchors [10200, 10710)
// Both boundaries and A_TOTAL are even -> an even-aligned anchor PAIR never
// straddles a level or the end of the anchor array.
#define A_TOTAL 10710
#define A_L0    8160
#define A_L01   10200
#define MAX_G   64
#define BLOCK   256
#define APT     2                 // anchors per thread (pair)
#define A_BLK   (BLOCK * APT)     // anchors per block

typedef float v4f __attribute__((ext_vector_type(4)));
typedef float v2f __attribute__((ext_vector_type(2)));

__global__ __launch_bounds__(BLOCK) void yolox_decode_assign_kernel(
    const float* __restrict__ preds,    // (B, A, 16)
    const float* __restrict__ labels,   // (B, G, 5)  [cls, cx, cy, w, h]
    float* __restrict__ out,            // decoded (B,A,16) | fg (B,A) | ibc (B,G,A)
    int B, int G)
{
    const int b   = blockIdx.y;
    const int tid = threadIdx.x;
    const int a0  = blockIdx.x * A_BLK + tid * APT;   // even

    __shared__ float sRaw[MAX_G * 5];   // raw labels for this batch
    __shared__ float sG[MAX_G][8];      // x0,x1,y0,y1,gx,gy,valid,pad

    // ---- Stage 1: async-copy labels[b] (G*5 floats) into LDS (gfx1250 async path) ----
    const int nLbl = G * 5;
    for (int i = tid; i < nLbl; i += BLOCK) {
        const float* gsrc = labels + (size_t)b * nLbl + i;
        uint64_t gaddr = (uint64_t)(uintptr_t)gsrc;
        uint32_t laddr = (uint32_t)(uintptr_t)(&sRaw[i]);   // low 32 bits = LDS offset
        asm volatile("global_load_async_to_lds_b32 %0, %1, off"
                     :: "v"(laddr), "v"(gaddr) : "memory");
    }
    asm volatile("s_wait_asynccnt 0" ::: "memory");
    __syncthreads();

    // ---- Stage 2: derive per-gt box data (one pass, then broadcast-read) ----
    if (tid < G) {
        float c  = sRaw[tid * 5 + 0];
        float gx = sRaw[tid * 5 + 1];
        float gy = sRaw[tid * 5 + 2];
        float gw = sRaw[tid * 5 + 3];
        float gh = sRaw[tid * 5 + 4];
        float v  = ((c + gx + gy + gw + gh) > 0.0f) ? 1.0f : 0.0f;  // valid[b,g]
        sG[tid][0] = gx - 0.5f * gw;
        sG[tid][1] = gx + 0.5f * gw;
        sG[tid][2] = gy - 0.5f * gh;
        sG[tid][3] = gy + 0.5f * gh;
        sG[tid][4] = gx;
        sG[tid][5] = gy;
        sG[tid][6] = v;
        sG[tid][7] = 0.0f;
    }
    __syncthreads();

    if (a0 >= A_TOTAL) return;   // pairs never straddle A_TOTAL (it is even)

    // ---- level is uniform across the pair (boundaries are even) ----
    int rel0, w; float s;
    if (a0 < A_L0)       { rel0 = a0;         w = 120; s = 8.0f;  }
    else if (a0 < A_L01) { rel0 = a0 - A_L0;  w = 60;  s = 16.0f; }
    else                 { rel0 = a0 - A_L01; w = 30;  s = 32.0f; }
    const int yg0 = rel0 / w;
    const int xg0 = rel0 - yg0 * w;

    // ---- decode both anchors (NT: every byte touched exactly once) ----
    float xc[APT], yc[APT];
    #pragma unroll
    for (int i = 0; i < APT; ++i) {
        int xg = xg0 + i, yg = yg0;
        if (xg >= w) { xg -= w; yg += 1; }          // row carry within level

        const size_t rowOff = ((size_t)b * A_TOTAL + (a0 + i)) * 16;
        const v4f* p4 = (const v4f*)(preds + rowOff);
        v4f q0 = __builtin_nontemporal_load(p4 + 0);
        v4f q1 = __builtin_nontemporal_load(p4 + 1);
        v4f q2 = __builtin_nontemporal_load(p4 + 2);
        v4f q3 = __builtin_nontemporal_load(p4 + 3);

        v4f d0;
        d0.x = (q0.x + (float)xg) * s;
        d0.y = (q0.y + (float)yg) * s;
        d0.z = __expf(q0.z) * s;
        d0.w = __expf(q0.w) * s;

        v4f* o4 = (v4f*)(out + rowOff);
        __builtin_nontemporal_store(d0, o4 + 0);
        __builtin_nontemporal_store(q1, o4 + 1);
        __builtin_nontemporal_store(q2, o4 + 2);
        __builtin_nontemporal_store(q3, o4 + 3);

        xc[i] = ((float)xg + 0.5f) * s;
        yc[i] = ((float)yg + 0.5f) * s;
    }

    // ---- containment tests: per-anchor 64-bit gt masks (G <= 64) ----
    const float r = 2.5f * s;
    uint64_t ib0 = 0, ic0 = 0, ib1 = 0, ic1 = 0;
    for (int g = 0; g < G; ++g) {
        v4f e0 = *(const v4f*)(&sG[g][0]);   // x0,x1,y0,y1 (ds_load_b128 broadcast)
        v4f e1 = *(const v4f*)(&sG[g][4]);   // gx,gy,valid,pad
        bool v  = e1.z > 0.0f;
        bool b0 = (xc[0] > e0.x) & (xc[0] < e0.y) & (yc[0] > e0.z) & (yc[0] < e0.w) & v;
        bool c0 = (fabsf(xc[0] - e1.x) < r) & (fabsf(yc[0] - e1.y) < r) & v;
        bool b1 = (xc[1] > e0.x) & (xc[1] < e0.y) & (yc[1] > e0.z) & (yc[1] < e0.w) & v;
        bool c1 = (fabsf(xc[1] - e1.x) < r) & (fabsf(yc[1] - e1.y) < r) & v;
        ib0 |= ((uint64_t)b0) << g;
        ic0 |= ((uint64_t)c0) << g;
        ib1 |= ((uint64_t)b1) << g;
        ic1 |= ((uint64_t)c1) << g;
    }

    const size_t fgBase  = (size_t)B * A_TOTAL * 16;
    const size_t ibcBase = fgBase + (size_t)B * A_TOTAL;

    // fg_mask[b, a0:a0+2] — one 8B NT store (all offsets even -> 8B aligned)
    v2f fgv;
    fgv.x = ((ib0 | ic0) != 0ull) ? 1.0f : 0.0f;
    fgv.y = ((ib1 | ic1) != 0ull) ? 1.0f : 0.0f;
    __builtin_nontemporal_store(fgv, (v2f*)(out + fgBase + (size_t)b * A_TOTAL + a0));

    // in_box_and_ctr[b,g,a0:a0+2]: fg-masking implied (fg==0 -> masks are 0).
    // Lane-contiguous v2f stores: each wave-store is a contiguous 256B burst.
    const uint64_t m0 = ib0 & ic0;
    const uint64_t m1 = ib1 & ic1;
    float* obase = out + ibcBase + (size_t)b * G * A_TOTAL + a0;
    for (int g = 0; g < G; ++g) {
        v2f val;
        val.x = (float)((m0 >> g) & 1ull);
        val.y = (float)((m1 >> g) & 1ull);
        __builtin_nontemporal_store(val, (v2f*)(obase + (size_t)g * A_TOTAL));
    }
}

extern "C" void kernel_launch(void* const* d_in, const int* in_sizes, int n_in,
                              void* d_out, int out_size, void* d_ws, size_t ws_size,
                              hipStream_t stream) {
    (void)n_in; (void)out_size; (void)d_ws; (void)ws_size;
    const float* preds  = (const float*)d_in[0];
    const float* labels = (const float*)d_in[1];
    float* out = (float*)d_out;

    const int A = A_TOTAL;
    const int B = in_sizes[0] / (A * 16);      // preds is (B, A, 16)
    int G = in_sizes[1] / (B * 5);             // labels is (B, G, 5)
    if (G > MAX_G) G = MAX_G;                  // bitmask path supports up to 64 gts

    dim3 grid((A + A_BLK - 1) / A_BLK, B);
    yolox_decode_assign_kernel<<<grid, BLOCK, 0, stream>>>(preds, labels, out, B, G);
}